// Sparse_attention_35459249996108
// MI455X (gfx1250) — compile-verified
//
#include <hip/hip_runtime.h>

// Sparse attention renorm: per-row top-6 threshold, clip, renormalize.
// Input/output: [16384, 2048] fp32. Memory-bound: ~268 MB total traffic
// -> ~11.5 us floor at 23.3 TB/s. One read + one write per element.
// Row staging into LDS uses the CDNA5 Tensor Data Mover (one 8 KB tile per
// block, TENSORcnt), falling back to per-lane async global->LDS (ASYNCcnt),
// falling back to plain b128 loads.

#define COLS   2048
#define TPB    128
#define VPT    16            // values per thread; TPB * VPT == COLS
#define NWAVES (TPB / 32)    // wave32 on gfx1250
#define EPSF   1e-07f

#if defined(__gfx1250__)
#if __has_builtin(__builtin_amdgcn_tensor_load_to_lds) && \
    __has_builtin(__builtin_amdgcn_s_wait_tensorcnt)
#define USE_TDM 1
#endif
#if __has_builtin(__builtin_amdgcn_global_load_async_to_lds_b128) && \
    __has_builtin(__builtin_amdgcn_s_wait_asynccnt)
#define HAVE_ASYNC 1
#endif
#endif

#if defined(USE_TDM) || defined(HAVE_ASYNC)
#define STAGE_LDS 1
typedef int v4i __attribute__((ext_vector_type(4)));
typedef unsigned int u32x4 __attribute__((ext_vector_type(4)));
typedef int i32x4 __attribute__((ext_vector_type(4)));
typedef int i32x8 __attribute__((ext_vector_type(8)));
typedef __attribute__((address_space(1))) v4i *gv4i_p;   // global (__device__*)
typedef __attribute__((address_space(3))) v4i *lv4i_p;   // LDS    (__shared__*)
typedef __attribute__((address_space(3))) void *lds_void_p;
#endif

// compare-exchange, max to lower index (descending order)
__device__ __forceinline__ void ce_desc(float &a, float &b) {
  const float hi = fmaxf(a, b);
  const float lo = fminf(a, b);
  a = hi;
  b = lo;
}

// Batcher odd-even mergesort, 8 elements, descending (19 CEs, branchless).
__device__ __forceinline__ void sort8_desc(float x[8]) {
  ce_desc(x[0], x[1]); ce_desc(x[2], x[3]); ce_desc(x[4], x[5]); ce_desc(x[6], x[7]);
  ce_desc(x[0], x[2]); ce_desc(x[1], x[3]); ce_desc(x[4], x[6]); ce_desc(x[5], x[7]);
  ce_desc(x[1], x[2]); ce_desc(x[5], x[6]);
  ce_desc(x[0], x[4]); ce_desc(x[1], x[5]); ce_desc(x[2], x[6]); ce_desc(x[3], x[7]);
  ce_desc(x[2], x[4]); ce_desc(x[3], x[5]);
  ce_desc(x[1], x[2]); ce_desc(x[3], x[4]); ce_desc(x[5], x[6]);
}

// a,b sorted descending; on return a = sorted descending top-8 of union(a,b).
// max(a[i], b[7-i]) is the upper half of the bitonic split of (a, reverse(b));
// it is itself bitonic -> 3-stage bitonic merge network sorts it.
__device__ __forceinline__ void merge_top8(float a[8], const float b[8]) {
  float m[8];
#pragma unroll
  for (int i = 0; i < 8; ++i) m[i] = fmaxf(a[i], b[7 - i]);
  ce_desc(m[0], m[4]); ce_desc(m[1], m[5]); ce_desc(m[2], m[6]); ce_desc(m[3], m[7]);
  ce_desc(m[0], m[2]); ce_desc(m[1], m[3]); ce_desc(m[4], m[6]); ce_desc(m[5], m[7]);
  ce_desc(m[0], m[1]); ce_desc(m[2], m[3]); ce_desc(m[4], m[5]); ce_desc(m[6], m[7]);
#pragma unroll
  for (int i = 0; i < 8; ++i) a[i] = m[i];
}

__global__ __launch_bounds__(TPB) void sparse_attn_topk_norm(
    const float *__restrict__ in, float *__restrict__ out, int rows) {
  const int row  = blockIdx.x;
  const int tid  = threadIdx.x;
  const int lane = tid & 31;
  const int wid  = tid >> 5;
  const size_t base = (size_t)row * COLS + (size_t)tid * VPT;

  __shared__ float red[NWAVES][8];
  __shared__ float sums[NWAVES];
  __shared__ float s_delta, s_inv;
#if defined(STAGE_LDS)
  __shared__ __align__(16) float rowbuf[COLS];
#endif

  // ---- stage the row into LDS / registers ----
  float x[VPT];
#if defined(USE_TDM)
  {
    if (wid == 0) {
      // Tensor DMA descriptor (D#): one 2048x1 tile of 4-byte elements.
      const unsigned long long ga =
          (unsigned long long)(const void *)(in + (size_t)row * COLS);
      const unsigned lds_base = (unsigned)(unsigned long long)(lds_void_p)rowbuf;
      u32x4 g0;
      g0[0] = 1u;                                            // count=1, is_restore=0
      g0[1] = lds_base;                                      // lds_addr
      g0[2] = (unsigned)(ga & 0xFFFFFFFFu);                  // global_addr[31:0]
      g0[3] = (unsigned)((ga >> 32) & 0x01FFFFFFu) | (2u << 30);  // addr[56:32]|type=2
      i32x8 g1;
      g1[0] = 0x00020000;                    // workgroup_mask=0, data_size=2 (4B)
      g1[1] = (int)((COLS & 0xFFFF) << 16);  // tensor_dim0[15:0] in [63:48]
      g1[2] = (1 << 16);                     // tensor_dim0 hi=0, tensor_dim1=1
      g1[3] = (int)((COLS & 0xFFFF) << 16);  // tile_dim0 = 2048
      g1[4] = 1;                             // tile_dim1 = 1, tile_dim2 = 0
      g1[5] = COLS;                          // tensor_dim0_stride = 2048
      g1[6] = 0;
      g1[7] = 0;
      i32x4 z4 = {0, 0, 0, 0};               // groups 2/3 unused (<=2D tensor)
      i32x8 z8 = {0, 0, 0, 0, 0, 0, 0, 0};   // trailing group: unused
      __builtin_amdgcn_tensor_load_to_lds(g0, g1, z4, z4, z8, 0);
      __builtin_amdgcn_s_wait_tensorcnt(0);
    }
    __syncthreads();  // publish LDS tile to all waves
  }
#elif defined(HAVE_ASYNC)
  {
    const int l = tid * VPT;
#pragma unroll
    for (int i = 0; i < VPT; i += 4) {
      __builtin_amdgcn_global_load_async_to_lds_b128(
          (gv4i_p)(in + base + i), (lv4i_p)(&rowbuf[l + i]), 0, 0);
    }
    __builtin_amdgcn_s_wait_asynccnt(0);
    __syncthreads();  // LDS fence: no load hoisting above the async wait
  }
#endif

#if defined(STAGE_LDS)
  {
    const int l = tid * VPT;
#pragma unroll
    for (int i = 0; i < VPT; i += 4) {
      const float4 v = *(const float4 *)(&rowbuf[l + i]);
      x[i] = v.x; x[i + 1] = v.y; x[i + 2] = v.z; x[i + 3] = v.w;
    }
  }
#else
#pragma unroll
  for (int i = 0; i < VPT; i += 4) {
    const float4 v = *(const float4 *)(in + base + i);
    x[i] = v.x; x[i + 1] = v.y; x[i + 2] = v.z; x[i + 3] = v.w;
  }
#endif

  // ---- per-thread sorted top-8 of its 16 values ----
  float a[8], b[8];
#pragma unroll
  for (int i = 0; i < 8; ++i) { a[i] = x[i]; b[i] = x[8 + i]; }
  sort8_desc(a);
  sort8_desc(b);
  merge_top8(a, b);

  // ---- wave32 butterfly: all lanes converge to the wave's top-8 ----
#pragma unroll
  for (int s = 1; s < 32; s <<= 1) {
    float p[8];
#pragma unroll
    for (int i = 0; i < 8; ++i) p[i] = __shfl_xor(a[i], s, 32);
    merge_top8(a, p);
  }
  if (lane == 0) {
#pragma unroll
    for (int i = 0; i < 8; ++i) red[wid][i] = a[i];
  }
  __syncthreads();

  // ---- cross-wave merge (4 lists), threshold = 6th largest + eps ----
  if (tid == 0) {
    float r[8];
#pragma unroll
    for (int i = 0; i < 8; ++i) r[i] = red[0][i];
    for (int w = 1; w < NWAVES; ++w) {
      float q[8];
#pragma unroll
      for (int i = 0; i < 8; ++i) q[i] = red[w][i];
      merge_top8(r, q);
    }
    s_delta = r[5] + EPSF;  // TOP_K = 6 -> index 5
  }
  __syncthreads();

  // ---- clip + row-sum + normalize ----
  const float delta = s_delta;
  float w[VPT];
  float lsum = 0.0f;
#pragma unroll
  for (int i = 0; i < VPT; ++i) {
    w[i] = fmaxf(x[i] - delta, 0.0f);
    lsum += w[i];
  }
#pragma unroll
  for (int s = 1; s < 32; s <<= 1) lsum += __shfl_xor(lsum, s, 32);
  if (lane == 0) sums[wid] = lsum;
  __syncthreads();
  if (tid == 0) {
    float tot = EPSF;
    for (int wv = 0; wv < NWAVES; ++wv) tot += sums[wv];
    s_inv = 1.0f / tot;
  }
  __syncthreads();

  const float inv = s_inv;
#pragma unroll
  for (int i = 0; i < VPT; i += 4) {
    float4 o;
    o.x = w[i] * inv; o.y = w[i + 1] * inv;
    o.z = w[i + 2] * inv; o.w = w[i + 3] * inv;
    *(float4 *)(out + base + i) = o;
  }
  (void)rows;
}

extern "C" void kernel_launch(void *const *d_in, const int *in_sizes, int n_in,
                              void *d_out, int out_size, void *d_ws, size_t ws_size,
                              hipStream_t stream) {
  (void)n_in; (void)out_size; (void)d_ws; (void)ws_size;
  const float *in = (const float *)d_in[0];
  float *out = (float *)d_out;
  const int rows = in_sizes[0] / COLS;  // 16384
  sparse_attn_topk_norm<<<rows, TPB, 0, stream>>>(in, out, rows);
}